// User_Graph_sample_10007273799798
// MI455X (gfx1250) — compile-verified
//
#include <hip/hip_runtime.h>
#include <stdint.h>

#define KNEI 20
#define DIM 64
#define ROWB 256                    // bytes per feature row (64 * f32)
#define BUFB (KNEI * ROWB)          // 5120 B: one user's staged rows
#define WPB 4                       // waves per block
#define BLOCKT (WPB * 32)
#define NBLOCKS 8192

typedef float f32x2 __attribute__((ext_vector_type(2)));

// Load one user's 20 indices+weights via scalar path, issue 10 async b128
// Global->LDS gathers (16 B per lane, half-wave h covers rows 2j+h).
__device__ __forceinline__ void issue_user(const float* feat, const int* graph,
                                           const float* wmat, int uu,
                                           const uint32_t (&dst)[KNEI / 2],
                                           int h, int q, float (&swout)[KNEI])
{
    const int uw = __builtin_amdgcn_readfirstlane(uu);
    const int*   gp = graph + (size_t)uw * KNEI;
    const float* wp = wmat  + (size_t)uw * KNEI;
    int sidx[KNEI];
#pragma unroll
    for (int k = 0; k < KNEI; ++k) { sidx[k] = gp[k]; swout[k] = wp[k]; }
#pragma unroll
    for (int j = 0; j < KNEI / 2; ++j) {
        const uint32_t row  = (uint32_t)(h ? sidx[2 * j + 1] : sidx[2 * j]);
        const uint32_t voff = row * (uint32_t)ROWB + (uint32_t)(q * 16);
        asm volatile("global_load_async_to_lds_b128 %0, %1, %2"
                     :: "v"(dst[j]), "v"(voff), "s"(feat) : "memory");
    }
}

// Reduce 20 staged rows with scalar weights; stream the result out with a
// non-temporal store so the output never displaces features1 in L2.
__device__ __forceinline__ void compute_store(const float* buf, const float (&sw)[KNEI],
                                              float* out, int uu, int lane)
{
    float ax = 0.f, ay = 0.f;
#pragma unroll
    for (int r = 0; r < KNEI; ++r) {
        const f32x2 v = *reinterpret_cast<const f32x2*>(buf + r * DIM + lane * 2);
        ax = fmaf(sw[r], v.x, ax);
        ay = fmaf(sw[r], v.y, ay);
    }
    f32x2 o; o.x = ax; o.y = ay;
    const uint32_t voff = (uint32_t)uu * (uint32_t)ROWB + (uint32_t)(lane * 8);
    asm volatile("global_store_b64 %0, %1, %2 th:TH_STORE_NT"
                 :: "v"(voff), "v"(o), "s"(out) : "memory");
}

__global__ __launch_bounds__(BLOCKT)
void user_graph_gather_wsum(const float* __restrict__ feat,
                            const int*   __restrict__ graph,
                            const float* __restrict__ wmat,
                            float* __restrict__ out,
                            int nuser)
{
    // double-buffered per-wave staging: 4 waves * 2 * 5120 B = 40 KB
    __shared__ __align__(16) float smem[WPB * 2 * (BUFB / 4)];

    const int lane = threadIdx.x & 31;
    const int wave = threadIdx.x >> 5;
    const int h    = lane >> 4;        // half-wave: issues rows 2j+h
    const int q    = lane & 15;        // 16-byte chunk within a 256-B row
    const int gwav = blockIdx.x * WPB + wave;
    const int nwav = gridDim.x * WPB;

    const uint32_t ldsWave =
        (uint32_t)(uintptr_t)(&smem[0]) + (uint32_t)(wave * 2 * BUFB);
    const float* const bufA = &smem[(size_t)wave * 2 * (BUFB / 4)];
    const float* const bufB = bufA + (BUFB / 4);

    // kernel-constant LDS destination addresses for the async gathers
    uint32_t dstA[KNEI / 2], dstB[KNEI / 2];
#pragma unroll
    for (int j = 0; j < KNEI / 2; ++j) {
        dstA[j] = ldsWave + (uint32_t)((2 * j + h) * ROWB + q * 16);
        dstB[j] = dstA[j] + (uint32_t)BUFB;
    }

    int u = gwav;
    if (u >= nuser) return;

    float swA[KNEI], swB[KNEI];
    issue_user(feat, graph, wmat, u, dstA, h, q, swA);

    while (true) {
        // ---- stage A: prefetch next into B, reduce A ----
        const int  u1 = u + nwav;
        const bool has1 = (u1 < nuser);
        if (has1) {
            issue_user(feat, graph, wmat, u1, dstB, h, q, swB);
            const int upf = u1 + nwav;
            if (upf < nuser) {
                __builtin_prefetch(graph + (size_t)upf * KNEI, 0, 1);
                __builtin_prefetch(wmat  + (size_t)upf * KNEI, 0, 1);
            }
            asm volatile("s_wait_asynccnt 0xa" ::: "memory"); // buffer A landed
        } else {
            asm volatile("s_wait_asynccnt 0x0" ::: "memory");
        }
        compute_store(bufA, swA, out, u, lane);
        if (!has1) break;

        // ---- stage B: prefetch next into A, reduce B ----
        const int  u2 = u1 + nwav;
        const bool has2 = (u2 < nuser);
        if (has2) {
            issue_user(feat, graph, wmat, u2, dstA, h, q, swA);
            const int upf = u2 + nwav;
            if (upf < nuser) {
                __builtin_prefetch(graph + (size_t)upf * KNEI, 0, 1);
                __builtin_prefetch(wmat  + (size_t)upf * KNEI, 0, 1);
            }
            asm volatile("s_wait_asynccnt 0xa" ::: "memory"); // buffer B landed
        } else {
            asm volatile("s_wait_asynccnt 0x0" ::: "memory");
        }
        compute_store(bufB, swB, out, u1, lane);
        if (!has2) break;

        u = u2;
    }
}

extern "C" void kernel_launch(void* const* d_in, const int* in_sizes, int n_in,
                              void* d_out, int out_size, void* d_ws, size_t ws_size,
                              hipStream_t stream)
{
    const float* feat  = (const float*)d_in[0];  // features1 [N,64]
    // d_in[1] (feattures2) is unused by the reference computation
    const int*   graph = (const int*)d_in[2];    // user_graph [N,20] (int32)
    const float* wmat  = (const float*)d_in[3];  // user_matrix [N,20]
    float* outp = (float*)d_out;                 // [N,64]

    const int nuser = in_sizes[0] / DIM;
    int blocks = (nuser + WPB - 1) / WPB;
    if (blocks > NBLOCKS) blocks = NBLOCKS;
    user_graph_gather_wsum<<<blocks, BLOCKT, 0, stream>>>(feat, graph, wmat, outp, nuser);
}